// EdgeVGAEEncoder_22110491640015
// MI455X (gfx1250) — compile-verified
//
#include <hip/hip_runtime.h>

#define N_NODES 50000
#define N_EDGES 640000
#define EDGE_DIM 64
#define HID 128
#define LAT 64
#define SLOPE 0.15f

typedef float v2f __attribute__((ext_vector_type(2)));
typedef float v8f __attribute__((ext_vector_type(8)));

__device__ __forceinline__ v8f wmma_f32_16x16x4(v2f a, v2f b, v8f c) {
  // D = A(16x4 f32) * B(4x16 f32) + C(16x16 f32)
  return __builtin_amdgcn_wmma_f32_16x16x4_f32(false, a, false, b, (short)0, c, false, false);
}

__device__ __forceinline__ void atomic_add_f32(float* p, float v) {
  __hip_atomic_fetch_add(p, v, __ATOMIC_RELAXED, __HIP_MEMORY_SCOPE_AGENT);
}

// ---------------- trivial elementwise kernels ----------------
__global__ void zero_kernel(float* __restrict__ p, int n) {
  for (int i = blockIdx.x * blockDim.x + threadIdx.x; i < n; i += gridDim.x * blockDim.x)
    p[i] = 0.f;
}
__global__ void leaky_kernel(float* __restrict__ p, int n) {
  for (int i = blockIdx.x * blockDim.x + threadIdx.x; i < n; i += gridDim.x * blockDim.x) {
    float v = p[i];
    p[i] = v >= 0.f ? v : SLOPE * v;
  }
}

// ---------------- fused edge-conv layer ----------------
// LDS layout (floats). Weights stored "paired-K": element (k,n) at
// base + ((k>>1)*PWN + n)*2 + (k&1)  -> one ds_load_b64 per B fragment.
#define PWN 132                         // padded N pitch (pairs) for 128-wide weights
#define PA3 260                         // A-tile row pitch (16 rows x 256 cols max)
#define OFF_W1 0                        // eW1: 32 pair-rows            ->  8448
#define OFF_W2 (OFF_W1 + 32 * PWN * 2)  // eW2: 64 pair-rows            -> 16896
#define OFF_NW (OFF_W2 + 64 * PWN * 2)  // nW : 128 pair-rows           -> 33792
#define OFF_B1 (OFF_NW + 128 * PWN * 2)
#define OFF_B2 (OFF_B1 + HID)
#define OFF_NB (OFF_B2 + HID)
#define OFF_AT (OFF_NB + HID)           // 4 waves x 16 x PA3
#define EDGE_LDS (OFF_AT + 4 * 16 * PA3)  // 76160 floats = 297.5 KB

__global__ __launch_bounds__(128) void edge_layer_kernel(
    const float* __restrict__ feat,   // [N_NODES, HID] node features (x or h1)
    const int* __restrict__ dst,      // [N_EDGES] destination node per edge
    const float* __restrict__ ea,     // [N_EDGES, EDGE_DIM]
    const float* __restrict__ W1, const float* __restrict__ b1,   // [64,128],[128]
    const float* __restrict__ W2, const float* __restrict__ b2,   // [128,128],[128]
    const float* __restrict__ NW, const float* __restrict__ nb,   // [256,128],[128]
    float* __restrict__ agg)          // [N_NODES, HID] accumulated via atomics
{
  __shared__ float sm[EDGE_LDS];
  const int tid = threadIdx.x;

  // ---- stage weights into LDS (paired-K layout) ----
  for (int i = tid; i < EDGE_DIM * HID; i += 128) {
    int k = i >> 7, n = i & 127;
    sm[OFF_W1 + ((k >> 1) * PWN + n) * 2 + (k & 1)] = W1[i];
  }
  for (int i = tid; i < HID * HID; i += 128) {
    int k = i >> 7, n = i & 127;
    sm[OFF_W2 + ((k >> 1) * PWN + n) * 2 + (k & 1)] = W2[i];
  }
  for (int i = tid; i < 2 * HID * HID; i += 128) {
    int k = i >> 7, n = i & 127;
    sm[OFF_NW + ((k >> 1) * PWN + n) * 2 + (k & 1)] = NW[i];
  }
  for (int i = tid; i < HID; i += 128) {
    sm[OFF_B1 + i] = b1[i];
    sm[OFF_B2 + i] = b2[i];
    sm[OFF_NB + i] = nb[i];
  }
  __syncthreads();

  const int lane = tid & 31;
  const int wave = tid >> 5;
  const int half = lane >> 4;   // 0: lanes 0-15, 1: lanes 16-31
  const int l16 = lane & 15;
  float* at = &sm[OFF_AT + wave * 16 * PA3];

  const int ntiles = N_EDGES / 16;  // 40000
  for (int t = blockIdx.x * 4 + wave; t < ntiles; t += gridDim.x * 4) {
    const int e0 = t * 16;

    // ---- load edge_attr tile (16x64, contiguous 4KB) into A cols [0,64) ----
    {
      const float4* src = (const float4*)(ea + (size_t)e0 * EDGE_DIM);
#pragma unroll
      for (int j = 0; j < 8; ++j) {
        int i4 = lane + 32 * j;          // 0..255 float4s
        float4 v = src[i4];
        int el = i4 * 4, r = el >> 6, cc = el & 63;
        *(float4*)&at[r * PA3 + cc] = v;
      }
    }
    int dv = dst[e0 + l16];  // lanes 16-31 mirror lanes 0-15

    v8f c[8];

    // ---- stage 1: t = leaky(ea @ W1 + b1)  -> A cols [64,192) ----
#pragma unroll
    for (int j = 0; j < 8; ++j) {
      float bv = sm[OFF_B1 + j * 16 + l16];
#pragma unroll
      for (int v = 0; v < 8; ++v) c[j][v] = bv;
    }
#pragma unroll 4
    for (int kb = 0; kb < EDGE_DIM; kb += 4) {
      v2f a = *(const v2f*)&at[l16 * PA3 + kb + 2 * half];
      int pr = (kb >> 1) + half;
#pragma unroll
      for (int j = 0; j < 8; ++j) {
        v2f b = *(const v2f*)&sm[OFF_W1 + (pr * PWN + j * 16 + l16) * 2];
        c[j] = wmma_f32_16x16x4(a, b, c[j]);
      }
    }
#pragma unroll
    for (int j = 0; j < 8; ++j)
#pragma unroll
      for (int v = 0; v < 8; ++v) {
        float x = c[j][v];
        x = x >= 0.f ? x : SLOPE * x;
        at[(v + 8 * half) * PA3 + 64 + j * 16 + l16] = x;
      }

    // ---- stage 2: emb = t @ W2 + b2  -> A cols [128,256) ----
#pragma unroll
    for (int j = 0; j < 8; ++j) {
      float bv = sm[OFF_B2 + j * 16 + l16];
#pragma unroll
      for (int v = 0; v < 8; ++v) c[j][v] = bv;
    }
#pragma unroll 4
    for (int kb = 0; kb < HID; kb += 4) {
      v2f a = *(const v2f*)&at[l16 * PA3 + 64 + kb + 2 * half];
      int pr = (kb >> 1) + half;
#pragma unroll
      for (int j = 0; j < 8; ++j) {
        v2f b = *(const v2f*)&sm[OFF_W2 + (pr * PWN + j * 16 + l16) * 2];
        c[j] = wmma_f32_16x16x4(a, b, c[j]);
      }
    }
#pragma unroll
    for (int j = 0; j < 8; ++j)
#pragma unroll
      for (int v = 0; v < 8; ++v)
        at[(v + 8 * half) * PA3 + 128 + j * 16 + l16] = c[j][v];

    // ---- gather feat[dst] rows -> A cols [0,128) (L2-resident table) ----
    for (int r = 0; r < 16; ++r) {
      int dr = __shfl(dv, r, 32);
      float4 v = *(const float4*)(feat + (size_t)dr * HID + lane * 4);
      *(float4*)&at[r * PA3 + lane * 4] = v;
    }

    // ---- stage 3: msg = [x_dst | emb] @ NW + nb; scatter-add ----
#pragma unroll
    for (int j = 0; j < 8; ++j) {
      float bv = sm[OFF_NB + j * 16 + l16];
#pragma unroll
      for (int v = 0; v < 8; ++v) c[j][v] = bv;
    }
#pragma unroll 4
    for (int kb = 0; kb < 2 * HID; kb += 4) {
      v2f a = *(const v2f*)&at[l16 * PA3 + kb + 2 * half];
      int pr = (kb >> 1) + half;
#pragma unroll
      for (int j = 0; j < 8; ++j) {
        v2f b = *(const v2f*)&sm[OFF_NW + (pr * PWN + j * 16 + l16) * 2];
        c[j] = wmma_f32_16x16x4(a, b, c[j]);
      }
    }
#pragma unroll
    for (int v = 0; v < 8; ++v) {
      int dm = __shfl(dv, v + 8 * half, 32);  // dst of edge row m = v + 8*half
      float* rowp = agg + (size_t)dm * HID;
#pragma unroll
      for (int j = 0; j < 8; ++j)
        atomic_add_f32(rowp + j * 16 + l16, c[j][v]);
    }
  }
}

// ---------------- mu / logvar heads ----------------
#define PW2 68    // padded N pitch (pairs) for 64-wide weights
#define PAF 132   // A-tile pitch, 16x128
#define OFF_MU 0
#define OFF_LV (OFF_MU + 64 * PW2 * 2)
#define OFF_MB (OFF_LV + 64 * PW2 * 2)
#define OFF_LB (OFF_MB + LAT)
#define OFF_AF (OFF_LB + LAT)
#define HEAD_LDS (OFF_AF + 4 * 16 * PAF)  // ~101.5 KB

__global__ __launch_bounds__(128) void head_kernel(
    const float* __restrict__ h,
    const float* __restrict__ muW, const float* __restrict__ mub,
    const float* __restrict__ lvW, const float* __restrict__ lvb,
    float* __restrict__ out)  // [mu (N*LAT) | logvar (N*LAT)]
{
  __shared__ float sm[HEAD_LDS];
  const int tid = threadIdx.x;
  for (int i = tid; i < HID * LAT; i += 128) {
    int k = i >> 6, n = i & 63;
    int o = ((k >> 1) * PW2 + n) * 2 + (k & 1);
    sm[OFF_MU + o] = muW[i];
    sm[OFF_LV + o] = lvW[i];
  }
  for (int i = tid; i < LAT; i += 128) {
    sm[OFF_MB + i] = mub[i];
    sm[OFF_LB + i] = lvb[i];
  }
  __syncthreads();

  const int lane = tid & 31, wave = tid >> 5;
  const int half = lane >> 4, l16 = lane & 15;
  float* at = &sm[OFF_AF + wave * 16 * PAF];

  const int ntiles = N_NODES / 16;  // 3125 exactly
  for (int t = blockIdx.x * 4 + wave; t < ntiles; t += gridDim.x * 4) {
    const int n0 = t * 16;
    const float4* src = (const float4*)(h + (size_t)n0 * HID);
#pragma unroll
    for (int j = 0; j < 16; ++j) {
      int i4 = lane + 32 * j;  // 0..511 float4s
      float4 v = src[i4];
      int el = i4 * 4, r = el >> 7, cc = el & 127;
      *(float4*)&at[r * PAF + cc] = v;
    }
    v8f c[8];  // c[0..3]=mu tiles, c[4..7]=logvar tiles
#pragma unroll
    for (int j = 0; j < 4; ++j) {
      float bm = sm[OFF_MB + j * 16 + l16];
      float bl = sm[OFF_LB + j * 16 + l16];
#pragma unroll
      for (int v = 0; v < 8; ++v) { c[j][v] = bm; c[4 + j][v] = bl; }
    }
#pragma unroll 4
    for (int kb = 0; kb < HID; kb += 4) {
      v2f a = *(const v2f*)&at[l16 * PAF + kb + 2 * half];
      int pr = (kb >> 1) + half;
#pragma unroll
      for (int j = 0; j < 4; ++j) {
        v2f bm = *(const v2f*)&sm[OFF_MU + (pr * PW2 + j * 16 + l16) * 2];
        c[j] = wmma_f32_16x16x4(a, bm, c[j]);
        v2f bl = *(const v2f*)&sm[OFF_LV + (pr * PW2 + j * 16 + l16) * 2];
        c[4 + j] = wmma_f32_16x16x4(a, bl, c[4 + j]);
      }
    }
#pragma unroll
    for (int v = 0; v < 8; ++v) {
      size_t node = (size_t)(n0 + v + 8 * half);
#pragma unroll
      for (int j = 0; j < 4; ++j) {
        out[node * LAT + j * 16 + l16] = c[j][v];
        out[(size_t)N_NODES * LAT + node * LAT + j * 16 + l16] = c[4 + j][v];
      }
    }
  }
}

extern "C" void kernel_launch(void* const* d_in, const int* in_sizes, int n_in,
                              void* d_out, int out_size, void* d_ws, size_t ws_size,
                              hipStream_t stream) {
  (void)in_sizes; (void)n_in; (void)out_size; (void)ws_size;

  const float* x    = (const float*)d_in[0];
  const int*   dst  = ((const int*)d_in[1]) + N_EDGES;  // edge_index row 1
  const float* ea   = (const float*)d_in[2];
  const float* e1W1 = (const float*)d_in[3];
  const float* e1b1 = (const float*)d_in[4];
  const float* e1W2 = (const float*)d_in[5];
  const float* e1b2 = (const float*)d_in[6];
  const float* n1W  = (const float*)d_in[7];
  const float* n1b  = (const float*)d_in[8];
  const float* e2W1 = (const float*)d_in[9];
  const float* e2b1 = (const float*)d_in[10];
  const float* e2W2 = (const float*)d_in[11];
  const float* e2b2 = (const float*)d_in[12];
  const float* n2W  = (const float*)d_in[13];
  const float* n2b  = (const float*)d_in[14];
  const float* muW  = (const float*)d_in[15];
  const float* mub  = (const float*)d_in[16];
  const float* lvW  = (const float*)d_in[17];
  const float* lvb  = (const float*)d_in[18];

  float* h1 = (float*)d_ws;
  float* h2 = h1 + (size_t)N_NODES * HID;
  const int nfeat = N_NODES * HID;

  zero_kernel<<<512, 256, 0, stream>>>(h1, nfeat);
  edge_layer_kernel<<<1250, 128, 0, stream>>>(x, dst, ea, e1W1, e1b1, e1W2, e1b2, n1W, n1b, h1);
  leaky_kernel<<<512, 256, 0, stream>>>(h1, nfeat);

  zero_kernel<<<512, 256, 0, stream>>>(h2, nfeat);
  edge_layer_kernel<<<1250, 128, 0, stream>>>(h1, dst, ea, e2W1, e2b1, e2W2, e2b2, n2W, n2b, h2);
  leaky_kernel<<<512, 256, 0, stream>>>(h2, nfeat);

  head_kernel<<<512, 128, 0, stream>>>(h2, muW, mub, lvW, lvb, (float*)d_out);
}